// GraphTemporalBlock_29437705847123
// MI455X (gfx1250) — compile-verified
//
#include <hip/hip_runtime.h>
#include <hip/hip_bf16.h>

typedef float v2f __attribute__((ext_vector_type(2)));
typedef float v8f __attribute__((ext_vector_type(8)));

#define DIM 64
#define BN_EPS 1e-5f

// ---------------------------------------------------------------------------
// 1) init: out = b (broadcast), deg = 1 (self loop), colAcc = 0
// ---------------------------------------------------------------------------
__global__ __launch_bounds__(256) void init_kernel(float* __restrict__ out,
                                                   const float* __restrict__ b,
                                                   float* __restrict__ deg,
                                                   float* __restrict__ colAcc,
                                                   int n) {
    int idx = blockIdx.x * blockDim.x + threadIdx.x;
    if (idx < n * DIM) out[idx] = b[idx & (DIM - 1)];
    if (idx < n) deg[idx] = 1.0f;           // self-loop contributes 1 to degree
    if (idx < 2 * DIM) colAcc[idx] = 0.0f;
}

// ---------------------------------------------------------------------------
// 2) degree accumulation over edges (atomic f32 adds)
// ---------------------------------------------------------------------------
__global__ __launch_bounds__(256) void degree_kernel(const int* __restrict__ dst,
                                                     float* __restrict__ deg, int E) {
    int e = blockIdx.x * blockDim.x + threadIdx.x;
    if (e < E) atomicAdd(&deg[dst[e]], 1.0f);
}

// ---------------------------------------------------------------------------
// 3) deg -> rsqrt(deg)  (deg >= 1 always due to self loops)
// ---------------------------------------------------------------------------
__global__ __launch_bounds__(256) void dinv_kernel(float* __restrict__ deg, int n) {
    int i = blockIdx.x * blockDim.x + threadIdx.x;
    if (i < n) deg[i] = rsqrtf(deg[i]);
}

// ---------------------------------------------------------------------------
// 4) h = x @ W with V_WMMA_F32_16X16X4_F32.
//    One wave computes a 16x64 row block (4 accumulators of 16x16, K-loop of 16).
//    W (64x64 f32 = 16KB) staged in LDS once per workgroup.
//    A layout (16x4 f32): lane L, vgpr v -> A[L%16][2*(L/16)+v]
//    B layout (4x16 f32): lane L, vgpr v -> B[2*(L/16)+v][L%16]
//    C/D layout (16x16 f32): vgpr r, lane L -> D[r + 8*(L/16)][L%16]
// ---------------------------------------------------------------------------
__global__ __launch_bounds__(256) void gemm_wmma_kernel(const float* __restrict__ x,
                                                        const float* __restrict__ W,
                                                        float* __restrict__ h, int n) {
    __shared__ float Ws[DIM * DIM];
    for (int i = threadIdx.x; i < DIM * DIM; i += 256) Ws[i] = W[i];
    __syncthreads();

    const int lane  = threadIdx.x & 31;
    const int wave  = threadIdx.x >> 5;
    const int m0    = (blockIdx.x * 8 + wave) * 16;
    if (m0 >= n) return;                      // wave-uniform: EXEC stays all-ones

    const int mlane = lane & 15;
    const int khalf = lane >> 4;              // 0 or 1

    v8f acc0 = {}, acc1 = {}, acc2 = {}, acc3 = {};

    const float* xrow = x + (size_t)(m0 + mlane) * DIM;

#pragma unroll
    for (int kt = 0; kt < 16; ++kt) {
        const int kb = kt * 4 + khalf * 2;
        const float2 av = *reinterpret_cast<const float2*>(xrow + kb);
        v2f a; a.x = av.x; a.y = av.y;

        const float* w0 = &Ws[kb * DIM];
        const float* w1 = &Ws[(kb + 1) * DIM];
        v2f b0, b1, b2, b3;
        b0.x = w0[ 0 + mlane]; b0.y = w1[ 0 + mlane];
        b1.x = w0[16 + mlane]; b1.y = w1[16 + mlane];
        b2.x = w0[32 + mlane]; b2.y = w1[32 + mlane];
        b3.x = w0[48 + mlane]; b3.y = w1[48 + mlane];

        acc0 = __builtin_amdgcn_wmma_f32_16x16x4_f32(false, a, false, b0, (short)0, acc0, false, false);
        acc1 = __builtin_amdgcn_wmma_f32_16x16x4_f32(false, a, false, b1, (short)0, acc1, false, false);
        acc2 = __builtin_amdgcn_wmma_f32_16x16x4_f32(false, a, false, b2, (short)0, acc2, false, false);
        acc3 = __builtin_amdgcn_wmma_f32_16x16x4_f32(false, a, false, b3, (short)0, acc3, false, false);
    }

    const int orow0 = m0 + khalf * 8;
#pragma unroll
    for (int r = 0; r < 8; ++r) {
        float* hp = h + (size_t)(orow0 + r) * DIM + mlane;
        hp[ 0] = acc0[r];
        hp[16] = acc1[r];
        hp[32] = acc2[r];
        hp[48] = acc3[r];
    }
}

// ---------------------------------------------------------------------------
// 5) self-loop contribution: out[i,:] += h[i,:] * dinv[i]^2  (runs before the
//    edge scatter; stream order means no atomics needed here)
// ---------------------------------------------------------------------------
__global__ __launch_bounds__(256) void selfloop_kernel(const float* __restrict__ h,
                                                       const float* __restrict__ dinv,
                                                       float* __restrict__ out, int n) {
    int idx = blockIdx.x * blockDim.x + threadIdx.x;
    if (idx >= n * DIM) return;
    int i = idx >> 6;
    float di = dinv[i];
    out[idx] += h[idx] * di * di;
}

// ---------------------------------------------------------------------------
// 6) edge scatter: out[dst,:] += h[src,:] * dinv[src]*dinv[dst]
//    16 threads per edge, float4 read per thread (coalesced 256B row read),
//    4 global_atomic_add_f32 per thread.
// ---------------------------------------------------------------------------
__global__ __launch_bounds__(256) void scatter_kernel(const float* __restrict__ h,
                                                      const int* __restrict__ src,
                                                      const int* __restrict__ dst,
                                                      const float* __restrict__ dinv,
                                                      float* __restrict__ out, int E) {
    int gid = blockIdx.x * blockDim.x + threadIdx.x;
    int e = gid >> 4;
    if (e >= E) return;
    int part = (gid & 15) * 4;
    int s = src[e], d = dst[e];
    float nrm = dinv[s] * dinv[d];
    const float4 v = *reinterpret_cast<const float4*>(h + (size_t)s * DIM + part);
    float* o = out + (size_t)d * DIM + part;
    atomicAdd(o + 0, v.x * nrm);
    atomicAdd(o + 1, v.y * nrm);
    atomicAdd(o + 2, v.z * nrm);
    atomicAdd(o + 3, v.w * nrm);
}

// ---------------------------------------------------------------------------
// 7) column sums / sums-of-squares (for BatchNorm batch statistics)
// ---------------------------------------------------------------------------
__global__ __launch_bounds__(256) void colstats_kernel(const float* __restrict__ out,
                                                       int n, float* __restrict__ colAcc) {
    __shared__ float shs[4][DIM];
    __shared__ float shq[4][DIM];
    const int d = threadIdx.x & (DIM - 1);
    const int g = threadIdx.x >> 6;           // 0..3
    float s = 0.f, q = 0.f;
    for (int row = blockIdx.x * 4 + g; row < n; row += gridDim.x * 4) {
        float v = out[(size_t)row * DIM + d];
        s += v; q += v * v;
    }
    shs[g][d] = s; shq[g][d] = q;
    __syncthreads();
    if (g == 0) {
        s = shs[0][d] + shs[1][d] + shs[2][d] + shs[3][d];
        q = shq[0][d] + shq[1][d] + shq[2][d] + shq[3][d];
        atomicAdd(&colAcc[d], s);
        atomicAdd(&colAcc[DIM + d], q);
    }
}

// ---------------------------------------------------------------------------
// 8) BN affine params: scale = gamma * rsqrt(var+eps), shift = beta - mean*scale
// ---------------------------------------------------------------------------
__global__ void params_kernel(const float* __restrict__ colAcc,
                              const float* __restrict__ gamma,
                              const float* __restrict__ beta,
                              float* __restrict__ params, int n) {
    int d = threadIdx.x;
    if (d < DIM) {
        float inv_n = 1.0f / (float)n;
        float mean = colAcc[d] * inv_n;
        float var  = colAcc[DIM + d] * inv_n - mean * mean;   // biased variance
        float scale = rsqrtf(var + BN_EPS) * gamma[d];
        params[d]       = scale;
        params[DIM + d] = beta[d] - mean * scale;
    }
}

// ---------------------------------------------------------------------------
// 9) y = relu(out * scale + shift), in place
// ---------------------------------------------------------------------------
__global__ __launch_bounds__(256) void bnrelu_kernel(float* __restrict__ out,
                                                     const float* __restrict__ params, int n) {
    int idx = blockIdx.x * blockDim.x + threadIdx.x;
    if (idx >= n * DIM) return;
    int d = idx & (DIM - 1);
    float v = out[idx] * params[d] + params[DIM + d];
    out[idx] = fmaxf(v, 0.0f);
}

// ---------------------------------------------------------------------------
// launcher
// ---------------------------------------------------------------------------
extern "C" void kernel_launch(void* const* d_in, const int* in_sizes, int n_in,
                              void* d_out, int out_size, void* d_ws, size_t ws_size,
                              hipStream_t stream) {
    const float* x     = (const float*)d_in[0];
    const int*   edge  = (const int*)d_in[1];   // [2, E] row-major
    const float* W     = (const float*)d_in[2];
    const float* b     = (const float*)d_in[3];
    const float* gamma = (const float*)d_in[4];
    const float* beta  = (const float*)d_in[5];

    const int n = in_sizes[0] / DIM;
    const int E = in_sizes[1] / 2;
    const int* src = edge;
    const int* dst = edge + E;

    float* out = (float*)d_out;

    // workspace layout: h[n*64] | deg[n] | colAcc[128] | params[128]
    float* h      = (float*)d_ws;
    float* deg    = h + (size_t)n * DIM;
    float* colAcc = deg + n;
    float* params = colAcc + 2 * DIM;

    const int nd       = n * DIM;
    const int blk      = 256;
    const int grid_nd  = (nd + blk - 1) / blk;
    const int grid_n   = (n + blk - 1) / blk;
    const int grid_e   = (E + blk - 1) / blk;
    const int tiles    = (n + 15) / 16;
    const int grid_gm  = (tiles + 7) / 8;
    const int grid_sc  = (E * 16 + blk - 1) / blk;   // 16 threads per edge

    init_kernel<<<grid_nd, blk, 0, stream>>>(out, b, deg, colAcc, n);
    degree_kernel<<<grid_e, blk, 0, stream>>>(dst, deg, E);
    dinv_kernel<<<grid_n, blk, 0, stream>>>(deg, n);
    gemm_wmma_kernel<<<grid_gm, blk, 0, stream>>>(x, W, h, n);
    selfloop_kernel<<<grid_nd, blk, 0, stream>>>(h, deg, out, n);
    scatter_kernel<<<grid_sc, blk, 0, stream>>>(h, src, dst, deg, out, E);
    colstats_kernel<<<512, blk, 0, stream>>>(out, n, colAcc);
    params_kernel<<<1, DIM, 0, stream>>>(colAcc, gamma, beta, params, n);
    bnrelu_kernel<<<grid_nd, blk, 0, stream>>>(out, params, n);
}